// GNN_11785390260977
// MI455X (gfx1250) — compile-verified
//
#include <hip/hip_runtime.h>

#define N_NODES 100000
#define DHID    128
#define E_EDGES 1600000
#define BN_EPS  1e-5f

typedef __attribute__((ext_vector_type(2))) float v2f;
typedef __attribute__((ext_vector_type(8))) float v8f;

// ---------------------------------------------------------------- utilities
__global__ void k_zero(float* __restrict__ p, int n) {
  int i = blockIdx.x * blockDim.x + threadIdx.x;
  if (i < n) p[i] = 0.0f;
}

__global__ void k_degree(const int* __restrict__ dst, float* __restrict__ deg, int e) {
  int i = blockIdx.x * blockDim.x + threadIdx.x;
  if (i < e) unsafeAtomicAdd(&deg[dst[i]], 1.0f);
}

// deg (in-place) -> dinv; selfc = dinv*dinv   (deg includes the +1 self loop)
__global__ void k_dinv(float* __restrict__ deg_dinv, float* __restrict__ selfc, int n) {
  int i = blockIdx.x * blockDim.x + threadIdx.x;
  if (i < n) {
    float d = __frsqrt_rn(deg_dinv[i] + 1.0f);
    deg_dinv[i] = d;
    selfc[i]    = d * d;
  }
}

// ------------------------------------------------------ fp32 WMMA GEMM (K=128)
// C[n,128] = T(A)[n,128] @ W[128,128]  (+bias, +relu epilogue optional)
// AMODE: 0 = identity, 1 = relu(a), 2 = relu(a)*bnScale[k] + bnShift[k]
// Compile-time specialization keeps the k-loop branch-free so the scheduler can
// pipeline global_load_b64 -> ds_load_2addr_b64 -> wmma x8 without SALU noise.
template <int AMODE, bool HAS_BIAS, bool EPI_RELU>
__global__ __launch_bounds__(256) void k_gemm128(
    const float* __restrict__ A, const float* __restrict__ W,
    float* __restrict__ C, int nrows,
    const float* __restrict__ bnScale, const float* __restrict__ bnShift,
    const float* __restrict__ bias)
{
  // Stage W into LDS with K-pairs adjacent: wl2[(k/2)*128 + n] = {W[k][n], W[k+1][n]}
  __shared__ float wl[DHID * DHID];
  const int tid = threadIdx.x;
  for (int i = tid; i < DHID * DHID; i += 256) {
    int k = i >> 7, n = i & 127;
    wl[(((k >> 1) << 7) + n) * 2 + (k & 1)] = W[i];
  }
  __syncthreads();

  const int wave  = tid >> 5;
  const int lane  = tid & 31;
  const int laneM = lane & 15;
  const int hi    = lane >> 4;          // half-wave selects K+0/1 vs K+2/3
  const int m0    = (blockIdx.x * 8 + wave) << 4;
  if (m0 >= nrows) return;              // N is a multiple of 16: tiles fully in/out

  const float2* __restrict__ wl2  = (const float2*)wl;
  const float*  __restrict__ arow = A + (size_t)(m0 + laneM) * DHID;

  v8f acc[8];
#pragma unroll
  for (int t = 0; t < 8; ++t) acc[t] = (v8f){};

  for (int k = 0; k < DHID; k += 4) {
    float2 av = *(const float2*)(arow + k + hi * 2);
    if (AMODE == 1) {
      av.x = fmaxf(av.x, 0.0f);
      av.y = fmaxf(av.y, 0.0f);
    } else if (AMODE == 2) {
      int c0 = k + hi * 2;
      av.x = fmaxf(av.x, 0.0f) * bnScale[c0]     + bnShift[c0];
      av.y = fmaxf(av.y, 0.0f) * bnScale[c0 + 1] + bnShift[c0 + 1];
    }
    v2f a; a[0] = av.x; a[1] = av.y;

    const float2* __restrict__ brow = wl2 + (((k >> 1) + hi) << 7);
#pragma unroll
    for (int t = 0; t < 8; ++t) {
      float2 bv = brow[t * 16 + laneM];
      v2f b; b[0] = bv.x; b[1] = bv.y;
      acc[t] = __builtin_amdgcn_wmma_f32_16x16x4_f32(
          false, a, false, b, (short)0, acc[t], false, false);
    }
  }

#pragma unroll
  for (int t = 0; t < 8; ++t) {
    int   col  = t * 16 + laneM;
    float bval = HAS_BIAS ? bias[col] : 0.0f;
#pragma unroll
    for (int r = 0; r < 8; ++r) {
      int   row = m0 + r + hi * 8;
      float v   = acc[t][r] + bval;
      if (EPI_RELU) v = fmaxf(v, 0.0f);
      C[(size_t)row * DHID + col] = v;
    }
  }
}

// out[i,c] = h[i,c]*selfc[i] + b[c]  (self-loop term + bias, pre-scatter init)
// float4-vectorized: one thread handles 4 channels.
__global__ void k_selfbias(const float* __restrict__ h, const float* __restrict__ selfc,
                           const float* __restrict__ b, float* __restrict__ out, int n) {
  int i = blockIdx.x * blockDim.x + threadIdx.x;       // over n*32 float4s
  if (i >= n * (DHID / 4)) return;
  int    node = i >> 5;
  int    c4   = (i & 31) * 4;
  float  sc   = selfc[node];
  float4 hv   = *(const float4*)(h + (size_t)node * DHID + c4);
  float4 bv   = *(const float4*)(b + c4);
  float4 o;
  o.x = hv.x * sc + bv.x;
  o.y = hv.y * sc + bv.y;
  o.z = hv.z * sc + bv.z;
  o.w = hv.w * sc + bv.w;
  *(float4*)(out + (size_t)node * DHID + c4) = o;
}

// one wave per edge: out[dst] += h[src] * (dinv[src]*dinv[dst]) ; L2-resident fp32 atomics
__global__ __launch_bounds__(256) void k_scatter(
    const int* __restrict__ src, const int* __restrict__ dst,
    const float* __restrict__ dinv, const float* __restrict__ h,
    float* __restrict__ out, int e)
{
  int w    = (blockIdx.x * blockDim.x + threadIdx.x) >> 5;
  int lane = threadIdx.x & 31;
  if (w >= e) return;
  int   s = src[w], d = dst[w];
  float c = dinv[s] * dinv[d];
  float4 hv = *(const float4*)(h + (size_t)s * DHID + lane * 4);
  float* o  = out + (size_t)d * DHID + lane * 4;
  unsafeAtomicAdd(o + 0, hv.x * c);
  unsafeAtomicAdd(o + 1, hv.y * c);
  unsafeAtomicAdd(o + 2, hv.z * c);
  unsafeAtomicAdd(o + 3, hv.w * c);
}

// per-channel sum / sumsq of relu(h) -> stats[0..127]=sum, stats[128..255]=sumsq
__global__ __launch_bounds__(256) void k_bnstats(const float* __restrict__ h,
                                                 float* __restrict__ stats, int n) {
  __shared__ float ss[256], sq[256];
  int t = threadIdx.x, c = t & 127, half = t >> 7;
  float s = 0.0f, q = 0.0f;
  for (int r = blockIdx.x * 2 + half; r < n; r += gridDim.x * 2) {
    float v = fmaxf(h[(size_t)r * DHID + c], 0.0f);
    s += v; q += v * v;
  }
  ss[t] = s; sq[t] = q;
  __syncthreads();
  if (t < 128) {
    unsafeAtomicAdd(&stats[c],       ss[t] + ss[t + 128]);
    unsafeAtomicAdd(&stats[128 + c], sq[t] + sq[t + 128]);
  }
}

__global__ void k_bnfinal(const float* __restrict__ stats, const float* __restrict__ gamma,
                          const float* __restrict__ beta, float* __restrict__ bn, int n) {
  int c = threadIdx.x;
  if (c < 128) {
    float inv = 1.0f / (float)n;
    float mu  = stats[c] * inv;
    float var = stats[128 + c] * inv - mu * mu;
    float sc  = gamma[c] * __frsqrt_rn(var + BN_EPS);
    bn[c]       = sc;                    // scale
    bn[128 + c] = beta[c] - mu * sc;     // shift
  }
}

// head: out[n,0:2] = hc[n,:] @ Wr + br   (hc already has +bc and relu applied)
__global__ __launch_bounds__(256) void k_head(const float* __restrict__ hc,
                                              const float* __restrict__ Wr,
                                              const float* __restrict__ br,
                                              float* __restrict__ out, int n) {
  int w    = (blockIdx.x * blockDim.x + threadIdx.x) >> 5;
  int lane = threadIdx.x & 31;
  if (w >= n) return;
  float4 v = *(const float4*)(hc + (size_t)w * DHID + lane * 4);
  int c = lane * 4;
  float p0 = v.x * Wr[c * 2]       + v.y * Wr[(c + 1) * 2]
           + v.z * Wr[(c + 2) * 2] + v.w * Wr[(c + 3) * 2];
  float p1 = v.x * Wr[c * 2 + 1]       + v.y * Wr[(c + 1) * 2 + 1]
           + v.z * Wr[(c + 2) * 2 + 1] + v.w * Wr[(c + 3) * 2 + 1];
#pragma unroll
  for (int off = 16; off > 0; off >>= 1) {
    p0 += __shfl_down(p0, off, 32);
    p1 += __shfl_down(p1, off, 32);
  }
  if (lane == 0) {
    out[(size_t)w * 2]     = p0 + br[0];
    out[(size_t)w * 2 + 1] = p1 + br[1];
  }
}

// ---------------------------------------------------------------- launcher
extern "C" void kernel_launch(void* const* d_in, const int* in_sizes, int n_in,
                              void* d_out, int out_size, void* d_ws, size_t ws_size,
                              hipStream_t stream) {
  const float* x     = (const float*)d_in[0];
  const int*   eidx  = (const int*)d_in[1];
  const float* W1    = (const float*)d_in[2];
  const float* b1    = (const float*)d_in[3];
  const float* W2    = (const float*)d_in[4];
  const float* b2    = (const float*)d_in[5];
  const float* W3    = (const float*)d_in[6];
  const float* b3    = (const float*)d_in[7];
  const float* gamma = (const float*)d_in[8];
  const float* beta  = (const float*)d_in[9];
  const float* Wc    = (const float*)d_in[10];
  const float* bc    = (const float*)d_in[11];
  const float* Wr    = (const float*)d_in[12];
  const float* br    = (const float*)d_in[13];
  float* out = (float*)d_out;

  const int N = N_NODES, E = E_EDGES;
  const int* src = eidx;
  const int* dst = eidx + E;

  float* buf0  = (float*)d_ws;                 // [N,128]
  float* buf1  = buf0 + (size_t)N * DHID;      // [N,128]
  float* dinv  = buf1 + (size_t)N * DHID;      // [N]   (deg -> dinv in place)
  float* selfc = dinv + N;                     // [N]
  float* stats = selfc + N;                    // [256]
  float* bnp   = stats + 256;                  // [256] scale|shift

  const dim3 blk(256);
  const int gN    = (N + 255) / 256;
  const int gE    = (E + 255) / 256;
  const int gNF4  = (N * (DHID / 4) + 255) / 256;
  const int gGemm = (N + 127) / 128;           // 8 waves * 16 rows per block
  const int gScat = (E * 32 + 255) / 256;      // one wave per edge
  const int gHead = (N * 32 + 255) / 256;      // one wave per node

  // degrees / normalization coefficients
  k_zero<<<gN, blk, 0, stream>>>(dinv, N);
  k_zero<<<1, blk, 0, stream>>>(stats, 256);
  k_degree<<<gE, blk, 0, stream>>>(dst, dinv, E);
  k_dinv<<<gN, blk, 0, stream>>>(dinv, selfc, N);

  // layer 1
  k_gemm128<0, false, false><<<gGemm, blk, 0, stream>>>(x, W1, buf0, N, nullptr, nullptr, nullptr);
  k_selfbias<<<gNF4, blk, 0, stream>>>(buf0, selfc, b1, buf1, N);
  k_scatter<<<gScat, blk, 0, stream>>>(src, dst, dinv, buf0, buf1, E);

  // layer 2 (relu fused into A-load)
  k_gemm128<1, false, false><<<gGemm, blk, 0, stream>>>(buf1, W2, buf0, N, nullptr, nullptr, nullptr);
  k_selfbias<<<gNF4, blk, 0, stream>>>(buf0, selfc, b2, buf1, N);
  k_scatter<<<gScat, blk, 0, stream>>>(src, dst, dinv, buf0, buf1, E);

  // layer 3
  k_gemm128<1, false, false><<<gGemm, blk, 0, stream>>>(buf1, W3, buf0, N, nullptr, nullptr, nullptr);
  k_selfbias<<<gNF4, blk, 0, stream>>>(buf0, selfc, b3, buf1, N);
  k_scatter<<<gScat, blk, 0, stream>>>(src, dst, dinv, buf0, buf1, E);

  // batchnorm over relu(h3)
  k_bnstats<<<1024, blk, 0, stream>>>(buf1, stats, N);
  k_bnfinal<<<1, 128, 0, stream>>>(stats, gamma, beta, bnp, N);

  // classifier: A = bn(relu(h3)); epilogue +bc, relu
  k_gemm128<2, true, true><<<gGemm, blk, 0, stream>>>(buf1, Wc, buf0, N, bnp, bnp + 128, bc);

  // regression head
  k_head<<<gHead, blk, 0, stream>>>(buf0, Wr, br, out, N);
}